// GNNGuard_82471962018398
// MI455X (gfx1250) — compile-verified
//
#include <hip/hip_runtime.h>
#include <math.h>

typedef __attribute__((ext_vector_type(2))) float v2f;
typedef __attribute__((ext_vector_type(8))) float v8f;

#define P0_THRESH 0.5f

// ---------------------------------------------------------------------------
// 1) Row L2-normalize: one wave per node.
__global__ __launch_bounds__(256) void rownorm_kernel(
    const float* __restrict__ x, float* __restrict__ xn, int n, int F) {
  int lane = threadIdx.x & 31;
  int i = blockIdx.x * (blockDim.x >> 5) + (threadIdx.x >> 5);
  if (i >= n) return;
  const float* xi = x + (long)i * F;
  float acc = 0.f;
  for (int f = lane; f < F; f += 32) { float v = xi[f]; acc += v * v; }
  for (int off = 16; off > 0; off >>= 1) acc += __shfl_down(acc, off);
  acc = __shfl(acc, 0);
  float nrm = sqrtf(acc);
  float inv = (nrm == 0.f) ? 1.0f : 1.0f / nrm;
  float* xo = xn + (long)i * F;
  for (int f = lane; f < F; f += 32) xo[f] = xi[f] * inv;
}

// ---------------------------------------------------------------------------
// 2) Edge cosine sim + threshold + diagonal kill; segment-sum into rs / degcnt.
__global__ __launch_bounds__(256) void edge_sim_kernel(
    const float* __restrict__ xn, const int* __restrict__ row,
    const int* __restrict__ col, float* __restrict__ sim,
    float* __restrict__ rs, float* __restrict__ degcnt, int E, int F) {
  int lane = threadIdx.x & 31;
  int e = blockIdx.x * (blockDim.x >> 5) + (threadIdx.x >> 5);
  if (e >= E) return;
  int r = row[e], c = col[e];
  const float* xr = xn + (long)r * F;
  const float* xc = xn + (long)c * F;
  float acc = 0.f;
  for (int f = lane; f < F; f += 32) acc += xr[f] * xc[f];
  for (int off = 16; off > 0; off >>= 1) acc += __shfl_down(acc, off);
  if (lane == 0) {
    float s = acc;
    if (s < P0_THRESH) s = 0.f;
    if (r == c) s = 0.f;
    sim[e] = s;
    if (s > 0.f) {
      atomicAdd(&rs[r], s);
      atomicAdd(&degcnt[r], 1.0f);
    }
  }
}

// ---------------------------------------------------------------------------
// 3) Edge weight: w = exp(sim/rs[row]) for kept edges, exp(1/(deg+1)) for the
//    tail self-loops. Also accumulate gcn degree = segment_sum(w, col).
__global__ __launch_bounds__(256) void edge_w_kernel(
    const float* __restrict__ sim, const int* __restrict__ row,
    const int* __restrict__ col, const float* __restrict__ rs,
    const float* __restrict__ degcnt, float* __restrict__ w,
    float* __restrict__ degw, int E, int nNodes) {
  int e = blockIdx.x * blockDim.x + threadIdx.x;
  if (e >= E) return;
  int r = row[e];
  float wv;
  if (e >= E - nNodes) {            // tail self-loops
    wv = expf(1.0f / (degcnt[r] + 1.0f));
  } else {
    float s = sim[e];
    wv = (s > 0.f) ? expf(s / rs[r]) : 0.f;
  }
  w[e] = wv;
  if (wv != 0.f) atomicAdd(&degw[col[e]], wv);
}

// ---------------------------------------------------------------------------
// 4) dis = deg>0 ? rsqrt(deg) : 0
__global__ __launch_bounds__(256) void dis_kernel(
    const float* __restrict__ degw, float* __restrict__ dis, int n) {
  int i = blockIdx.x * blockDim.x + threadIdx.x;
  if (i >= n) return;
  float d = degw[i];
  dis[i] = (d > 0.f) ? rsqrtf(d) : 0.f;
}

// ---------------------------------------------------------------------------
// 5) fp32 WMMA GEMM: C[M x N] = A[M x K] * B[K x N], row-major.
//    B staged once per workgroup into LDS, packed as (k,k+1) float2 pairs so a
//    B fragment is a single ds_load_b64. Each wave owns a 16-row stripe and
//    TN=N/16 column tiles: one global_load_b64 of the A fragment feeds TN
//    v_wmma_f32_16x16x4_f32 ops (independent accumulators).
//    Fragment layouts per CDNA5 ISA 7.12.2 (lanes 0-15 ↔ K 0/1, 16-31 ↔ K 2/3).
template <int TN>
__global__ __launch_bounds__(256) void gemm_wmma_f32_kernel(
    const float* __restrict__ A, const float* __restrict__ B,
    float* __restrict__ C, int M, int K, int N) {
  __shared__ v2f Bs[64 * 128];             // up to (K/2=64) x (N=128) = 64 KB
  int tid = threadIdx.x;
  int pairs = (K >> 1) * N;
  for (int idx = tid; idx < pairs; idx += 256) {
    int kk = idx / N, n2 = idx - kk * N;
    v2f b;
    b.x = B[(long)(2 * kk) * N + n2];
    b.y = B[(long)(2 * kk + 1) * N + n2];
    Bs[idx] = b;
  }
  __syncthreads();

  int lane = tid & 31;
  int wave = tid >> 5;
  int row0 = (blockIdx.x * 8 + wave) << 4;
  if (row0 >= M) return;                   // wave-uniform: EXEC all-1s inside
  int half = lane >> 4;                    // 0 or 1
  int l15 = lane & 15;
  int kh = half << 1;                      // +0 or +2 within the K=4 slab
  const float* Arow = A + (long)(row0 + l15) * K;

  v8f acc[TN];
#pragma unroll
  for (int t = 0; t < TN; ++t) acc[t] = (v8f){};

  for (int k = 0; k < K; k += 4) {
    v2f a = *(const v2f*)(Arow + k + kh);              // global_load_b64
    const v2f* brow = &Bs[((k >> 1) + half) * N];
#pragma unroll
    for (int t = 0; t < TN; ++t) {
      v2f b = brow[(t << 4) + l15];                    // ds_load_b64
      acc[t] = __builtin_amdgcn_wmma_f32_16x16x4_f32(
          /*neg_a=*/false, a, /*neg_b=*/false, b,
          /*c_mod=*/(short)0, acc[t], /*reuse_a=*/false, /*reuse_b=*/false);
    }
  }

  int mbase = row0 + (half << 3);
#pragma unroll
  for (int t = 0; t < TN; ++t) {
#pragma unroll
    for (int j = 0; j < 8; ++j)
      C[(long)(mbase + j) * N + (t << 4) + l15] = acc[t][j];
  }
}

// ---------------------------------------------------------------------------
// 6) out[i][f] = bias[f]
__global__ __launch_bounds__(256) void init_out_kernel(
    float* __restrict__ out, const float* __restrict__ bias, int n, int F) {
  long t = (long)blockIdx.x * blockDim.x + threadIdx.x;
  if (t >= (long)n * F) return;
  out[t] = bias[t % F];
}

// ---------------------------------------------------------------------------
// 7) out[col] += dis[row]*w*dis[col] * h[row] — one wave per edge.
__global__ __launch_bounds__(256) void aggregate_kernel(
    const float* __restrict__ h, const int* __restrict__ row,
    const int* __restrict__ col, const float* __restrict__ w,
    const float* __restrict__ dis, float* __restrict__ out, int E, int F) {
  int lane = threadIdx.x & 31;
  int e = blockIdx.x * (blockDim.x >> 5) + (threadIdx.x >> 5);
  if (e >= E) return;
  int r = row[e], c = col[e];
  float coef = dis[r] * w[e] * dis[c];
  if (coef == 0.f) return;                 // wave-uniform (scalar per edge)
  const float* hr = h + (long)r * F;
  float* oc = out + (long)c * F;
  for (int f = lane; f < F; f += 32) atomicAdd(&oc[f], coef * hr[f]);
}

// ---------------------------------------------------------------------------
// 8) ReLU in place.
__global__ __launch_bounds__(256) void relu_kernel(float* __restrict__ v, long n) {
  long t = (long)blockIdx.x * blockDim.x + threadIdx.x;
  if (t >= n) return;
  float x = v[t];
  v[t] = x > 0.f ? x : 0.f;
}

// ---------------------------------------------------------------------------
// 9) log_softmax over C classes, in place; one thread per node.
__global__ __launch_bounds__(256) void logsoftmax_kernel(
    float* __restrict__ o, int n, int C) {
  int i = blockIdx.x * blockDim.x + threadIdx.x;
  if (i >= n) return;
  float* oi = o + (long)i * C;
  float m = -INFINITY;
  for (int j = 0; j < C; ++j) m = fmaxf(m, oi[j]);
  float s = 0.f;
  for (int j = 0; j < C; ++j) s += expf(oi[j] - m);
  float l = logf(s) + m;
  for (int j = 0; j < C; ++j) oi[j] = oi[j] - l;
}

// ---------------------------------------------------------------------------
static inline int cdiv(long a, long b) { return (int)((a + b - 1) / b); }

static void run_layer(const float* feat, int Fin, int Fout,
                      const float* Wt, const float* bias,
                      const int* row, const int* col,
                      float* out, bool relu, int N, int E,
                      float* xn, float* hw, float* sim, float* w,
                      float* rs, float* degcnt, float* degw, float* dis,
                      hipStream_t stream) {
  hipMemsetAsync(rs, 0, (size_t)N * sizeof(float), stream);
  hipMemsetAsync(degcnt, 0, (size_t)N * sizeof(float), stream);
  hipMemsetAsync(degw, 0, (size_t)N * sizeof(float), stream);

  const int TB = 256, WPB = TB / 32;
  rownorm_kernel<<<cdiv(N, WPB), TB, 0, stream>>>(feat, xn, N, Fin);
  edge_sim_kernel<<<cdiv(E, WPB), TB, 0, stream>>>(xn, row, col, sim, rs, degcnt, E, Fin);
  edge_w_kernel<<<cdiv(E, TB), TB, 0, stream>>>(sim, row, col, rs, degcnt, w, degw, E, N);
  dis_kernel<<<cdiv(N, TB), TB, 0, stream>>>(degw, dis, N);

  int rowTiles = N >> 4;                   // N divisible by 16 (50000 = 16*3125)
  int blocks = cdiv(rowTiles, WPB);
  if (Fout == 128) {
    gemm_wmma_f32_kernel<8><<<blocks, TB, 0, stream>>>(feat, Wt, hw, N, Fin, Fout);
  } else {
    gemm_wmma_f32_kernel<1><<<blocks, TB, 0, stream>>>(feat, Wt, hw, N, Fin, Fout);
  }

  init_out_kernel<<<cdiv((long)N * Fout, TB), TB, 0, stream>>>(out, bias, N, Fout);
  aggregate_kernel<<<cdiv(E, WPB), TB, 0, stream>>>(hw, row, col, w, dis, out, E, Fout);
  if (relu) relu_kernel<<<cdiv((long)N * Fout, TB), TB, 0, stream>>>(out, (long)N * Fout);
}

extern "C" void kernel_launch(void* const* d_in, const int* in_sizes, int n_in,
                              void* d_out, int out_size, void* d_ws, size_t ws_size,
                              hipStream_t stream) {
  (void)n_in; (void)out_size; (void)ws_size;
  const float* x  = (const float*)d_in[0];
  const int* row  = (const int*)d_in[1];
  const int* col  = (const int*)d_in[2];
  const float* W0 = (const float*)d_in[3];
  const float* b0 = (const float*)d_in[4];
  const float* W1 = (const float*)d_in[5];
  const float* b1 = (const float*)d_in[6];
  const float* W2 = (const float*)d_in[7];
  const float* b2 = (const float*)d_in[8];

  const int NHID   = in_sizes[4];          // 128
  const int D2     = in_sizes[6];          // 16
  const int NCLASS = in_sizes[8];          // 16
  const int NFEAT  = in_sizes[3] / NHID;   // 128
  const int N      = in_sizes[0] / NFEAT;  // 50000
  const int E      = in_sizes[1];          // 850000

  // Workspace carve-up (floats).
  float* p = (float*)d_ws;
  float* xn     = p; p += (long)N * NFEAT;   // max Fin = 128
  float* hw     = p; p += (long)N * NHID;    // max Fout = 128
  float* h1     = p; p += (long)N * NHID;
  float* h2     = p; p += (long)N * D2;
  float* sim    = p; p += E;
  float* w      = p; p += E;
  float* rs     = p; p += N;
  float* degcnt = p; p += N;
  float* degw   = p; p += N;
  float* dis    = p; p += N;

  float* out = (float*)d_out;

  // Layer 0: 128 -> 128, ReLU
  run_layer(x,  NFEAT, NHID,   W0, b0, row, col, h1, true,  N, E,
            xn, hw, sim, w, rs, degcnt, degw, dis, stream);
  // Layer 1: 128 -> 16, ReLU
  run_layer(h1, NHID,  D2,     W1, b1, row, col, h2, true,  N, E,
            xn, hw, sim, w, rs, degcnt, degw, dis, stream);
  // Layer 2: 16 -> 16, no ReLU, straight into d_out
  run_layer(h2, D2,    NCLASS, W2, b2, row, col, out, false, N, E,
            xn, hw, sim, w, rs, degcnt, degw, dis, stream);

  logsoftmax_kernel<<<cdiv(N, 256), 256, 0, stream>>>(out, N, NCLASS);
}